// stacked_bi_36344013258827
// MI455X (gfx1250) — compile-verified
//
#include <hip/hip_runtime.h>
#include <hip/hip_bf16.h>

// ---------------------------------------------------------------------------
// Bidirectional long conv (SSM / FlashFFTConv style) on gfx1250.
//
// Math: y[b,c,t] = sum_{τ<=t} kf[c,τ] x[b,c,t-τ]
//                + sum_{τ<=L-1-t} kb[c,τ] x[b,c,t+τ]
//                + D[c] x[b,c,t]
// Implemented as ONE circular conv of length N=8192:
//   y = Re( IFFT( FFT(pad(x)) * G ) )[0:L] + D*x,  G = Kf + conj(Kb)
// Two batch elements are packed per complex FFT (x_b1 + i*x_b2).
//
// FFT: Stockham autosort, 3 radix-16 stages (via V_WMMA_F32_16X16X4_F32)
// + 1 radix-2 stage. One 8192-pt complex signal per workgroup in LDS.
// ---------------------------------------------------------------------------

typedef __attribute__((ext_vector_type(2))) float v2f;
typedef __attribute__((ext_vector_type(8))) float v8f;

#define NFFT   8192
#define LSEQ   4096
#define NCH    512
#define NBATCH 16
#define BLOCK  256
#define TWO_PI 6.2831853071795864769f

__device__ __forceinline__ v8f wmma_f32(v2f a, v2f b, v8f c) {
  // D(16x16,f32) += A(16x4,f32) x B(4x16,f32)
  // (neg_a, A, neg_b, B, c_mod, C, reuse_a, reuse_b) - mirrors confirmed f16 form
  return __builtin_amdgcn_wmma_f32_16x16x4_f32(false, a, false, b, (short)0, c,
                                               false, false);
}

// DFT-16 matrix in WMMA A-operand layout (32-bit A 16x4):
// lane l holds row M = l%16; VGPR v holds K = v + 2*(l/16).
// For K-chunk kk, r = 4*kk + K.  A[u][r] = exp(-2*pi*i*u*r/16).
__device__ __forceinline__ void make_dft16(int lane, v2f* Ar, v2f* Ai, v2f* An) {
  const int u = lane & 15;
  const int half = lane >> 4;
#pragma unroll
  for (int kk = 0; kk < 4; ++kk) {
#pragma unroll
    for (int v = 0; v < 2; ++v) {
      const int r = 4 * kk + 2 * half + v;
      const float th = -(float)((u * r) & 15) * (TWO_PI / 16.0f);
      const float cs = __cosf(th);
      const float sn = __sinf(th);
      Ar[kk][v] = cs;
      Ai[kk][v] = sn;
      An[kk][v] = -sn;  // -Im(D16) so "cr -= Ai*Bi" is an accumulate
    }
  }
}

// One Stockham radix-16 stage.  Current sub-length NSUB, stride S, m=NSUB/16.
// Column c in [0,512) encodes (p = c/S, q = c%S):
//   gather : in [ c + 512*r ],  r = 0..15              (S*m == 512 every stage)
//   DFT16 along r (WMMA), twiddle w = exp(-2πi*p*u/NSUB)
//   scatter: out[ q + 16*S*p + S*u ]
template <int S, int NSUB>
__device__ __forceinline__ void fft_stage(const float* __restrict__ inr,
                                          const float* __restrict__ ini,
                                          float* __restrict__ outr,
                                          float* __restrict__ outi, int tid,
                                          const v2f* Ar, const v2f* Ai,
                                          const v2f* An) {
  const int lane = tid & 31;
  const int wave = tid >> 5;   // 8 waves
  const int half = lane >> 4;
  const int col16 = lane & 15;

  for (int tile = wave; tile < 32; tile += 8) {  // 32 tiles of 16 columns
    const int col = tile * 16 + col16;
    v8f cr = {0.f, 0.f, 0.f, 0.f, 0.f, 0.f, 0.f, 0.f};
    v8f ci = {0.f, 0.f, 0.f, 0.f, 0.f, 0.f, 0.f, 0.f};
#pragma unroll
    for (int kk = 0; kk < 4; ++kk) {
      // B operand (4x16, 32-bit): lane holds col N=l%16, VGPR v -> K=v+2*(l/16)
      const int r0 = 4 * kk + 2 * half;
      v2f br, bi;
      br.x = inr[col + 512 * r0];
      br.y = inr[col + 512 * (r0 + 1)];
      bi.x = ini[col + 512 * r0];
      bi.y = ini[col + 512 * (r0 + 1)];
      // complex matmul: (cr,ci) += D16 * (br,bi)
      cr = wmma_f32(Ar[kk], br, cr);
      cr = wmma_f32(An[kk], bi, cr);
      ci = wmma_f32(Ar[kk], bi, ci);
      ci = wmma_f32(Ai[kk], br, ci);
    }
    // C/D layout: VGPR j holds row u = j + 8*(l/16) for column N = l%16.
    const int p = col / S;
    const int q = col - p * S;
    const int base = q + 16 * S * p;
#pragma unroll
    for (int j = 0; j < 8; ++j) {
      const int u = j + 8 * half;
      const float th = -(TWO_PI / (float)NSUB) * (float)(p * u);
      const float wr = __cosf(th);
      const float wi = __sinf(th);
      outr[base + S * u] = cr[j] * wr - ci[j] * wi;
      outi[base + S * u] = cr[j] * wi + ci[j] * wr;
    }
  }
}

// Forward 8192-pt complex FFT over LDS SoA buffers. Input & final result in
// (x0r,x0i); (x1r,x1i) is ping-pong scratch. Natural-order output.
__device__ __forceinline__ void fft8192(float* x0r, float* x0i, float* x1r,
                                        float* x1i, int tid, const v2f* Ar,
                                        const v2f* Ai, const v2f* An) {
  fft_stage<1, 8192>(x0r, x0i, x1r, x1i, tid, Ar, Ai, An);
  __syncthreads();
  fft_stage<16, 512>(x1r, x1i, x0r, x0i, tid, Ar, Ai, An);
  __syncthreads();
  fft_stage<256, 32>(x0r, x0i, x1r, x1i, tid, Ar, Ai, An);
  __syncthreads();
  // final radix-2: n=2, s=4096, twiddle = 1
  for (int q = tid; q < 4096; q += BLOCK) {
    const float ar = x1r[q], ai = x1i[q];
    const float br = x1r[q + 4096], bi = x1i[q + 4096];
    x0r[q] = ar + br;
    x0i[q] = ai + bi;
    x0r[q + 4096] = ar - br;
    x0i[q + 4096] = ai - bi;
  }
  __syncthreads();
}

// ---------------------------------------------------------------------------
// Kernel A: per output channel c, build G[c][f] = Kf[f] + conj(Kb[f]) using a
// single packed FFT of z = kf + i*kb:
//   G[f] = (Z[f]+conj(Z[-f]))/2 + (i/2)(conj(Z[f]) - Z[-f])
// ---------------------------------------------------------------------------
__global__ __launch_bounds__(BLOCK) void ssm_prep_kernel(
    const float* __restrict__ k, float2* __restrict__ G) {
  __shared__ float sm[4 * NFFT];
  float* x0r = sm;
  float* x0i = sm + NFFT;
  float* x1r = sm + 2 * NFFT;
  float* x1i = sm + 3 * NFFT;

  const int c = blockIdx.x;
  const int tid = threadIdx.x;
  const float* kf = k + (size_t)c * LSEQ;
  const float* kb = k + (size_t)(c + NCH) * LSEQ;

  for (int i = tid; i < LSEQ; i += BLOCK) {
    x0r[i] = kf[i];
    x0i[i] = kb[i];
  }
  for (int i = LSEQ + tid; i < NFFT; i += BLOCK) {
    x0r[i] = 0.f;
    x0i[i] = 0.f;
  }
  __syncthreads();

  v2f Ar[4], Ai[4], An[4];
  make_dft16(tid & 31, Ar, Ai, An);
  fft8192(x0r, x0i, x1r, x1i, tid, Ar, Ai, An);

  for (int f = tid; f < NFFT; f += BLOCK) {
    const int fm = (NFFT - f) & (NFFT - 1);
    const float zr = x0r[f], zi = x0i[f];
    const float zmr = x0r[fm], zmi = x0i[fm];
    const float gr = 0.5f * (zr + zmr + zi + zmi);
    const float gi = 0.5f * (zi - zmi + zr - zmr);
    G[(size_t)c * NFFT + f] = make_float2(gr, gi);
  }
}

// ---------------------------------------------------------------------------
// Kernel B: one (channel, batch-pair) per workgroup.
//   Z = FFT(pad(x_b1) + i*pad(x_b2));  W = Z*G;  F = FFT(conj(W))
//   y_b1 =  Re(F)/N + D*x_b1 ;  y_b2 = -Im(F)/N + D*x_b2
// ---------------------------------------------------------------------------
__global__ __launch_bounds__(BLOCK) void ssm_conv_kernel(
    const float* __restrict__ x, const float2* __restrict__ G,
    const float* __restrict__ D, float* __restrict__ out) {
  __shared__ float sm[4 * NFFT];
  float* x0r = sm;
  float* x0i = sm + NFFT;
  float* x1r = sm + 2 * NFFT;
  float* x1i = sm + 3 * NFFT;

  const int wg = blockIdx.x;
  const int c = wg & (NCH - 1);
  const int bp = wg >> 9;  // 0..7
  const int b1 = bp;
  const int b2 = bp + NBATCH / 2;
  const int tid = threadIdx.x;

  const float* x1p = x + ((size_t)b1 * NCH + c) * LSEQ;
  const float* x2p = x + ((size_t)b2 * NCH + c) * LSEQ;

  for (int i = tid; i < LSEQ; i += BLOCK) {
    x0r[i] = x1p[i];
    x0i[i] = x2p[i];
  }
  for (int i = LSEQ + tid; i < NFFT; i += BLOCK) {
    x0r[i] = 0.f;
    x0i[i] = 0.f;
  }
  __syncthreads();

  v2f Ar[4], Ai[4], An[4];
  make_dft16(tid & 31, Ar, Ai, An);

  fft8192(x0r, x0i, x1r, x1i, tid, Ar, Ai, An);  // Z in x0

  // W = Z*G, stored conjugated (inverse FFT via conj-forward-conj trick)
  const float2* Gc = G + (size_t)c * NFFT;
  for (int f = tid; f < NFFT; f += BLOCK) {
    const float2 g = Gc[f];
    const float zr = x0r[f], zi = x0i[f];
    x0r[f] = zr * g.x - zi * g.y;
    x0i[f] = -(zr * g.y + zi * g.x);
  }
  __syncthreads();

  fft8192(x0r, x0i, x1r, x1i, tid, Ar, Ai, An);  // F in x0

  const float Dc = D[c];
  const float scl = 1.0f / (float)NFFT;
  float* o1 = out + ((size_t)b1 * NCH + c) * LSEQ;
  float* o2 = out + ((size_t)b2 * NCH + c) * LSEQ;
  for (int t = tid; t < LSEQ; t += BLOCK) {
    o1[t] = x0r[t] * scl + Dc * x1p[t];
    o2[t] = -x0i[t] * scl + Dc * x2p[t];
  }
}

extern "C" void kernel_launch(void* const* d_in, const int* in_sizes, int n_in,
                              void* d_out, int out_size, void* d_ws,
                              size_t ws_size, hipStream_t stream) {
  (void)in_sizes;
  (void)n_in;
  (void)out_size;
  (void)ws_size;
  const float* x = (const float*)d_in[0];  // (16, 512, 4096) f32
  const float* k = (const float*)d_in[1];  // (1, 1024, 4096) f32
  const float* D = (const float*)d_in[2];  // (1, 512, 1)    f32
  float* out = (float*)d_out;              // (16, 512, 4096) f32
  float2* G = (float2*)d_ws;               // 512 * 8192 * 8 B = 32 MB

  ssm_prep_kernel<<<NCH, BLOCK, 0, stream>>>(k, G);
  ssm_conv_kernel<<<(NBATCH / 2) * NCH, BLOCK, 0, stream>>>(x, G, D, out);
}